// SpatialAttentionModule_44160853737545
// MI455X (gfx1250) — compile-verified
//
#include <hip/hip_runtime.h>
#include <math.h>

// Problem constants: x = [N=32, C=256, H=56, W=56] fp32
#define PN   32
#define PC   256
#define PH   56
#define PW   56
#define PHW  (PH * PW)          // 3136
#define PCHW (PC * PHW)         // 802816
#define PNHW (PN * PHW)         // 100352

typedef float v2f __attribute__((ext_vector_type(2)));
typedef float v8f __attribute__((ext_vector_type(8)));

// ---------------------------------------------------------------------------
// Kernel 1: channel-wise mean & max pooling.
// One thread handles 4 contiguous (h,w) pixels -> float4 loads (coalesced,
// 512B per wave per load). pooled layout: [N][2][H][W], ch0=avg, ch1=max.
// ---------------------------------------------------------------------------
__global__ void __launch_bounds__(256) sa_pool_kernel(
    const float* __restrict__ x, float* __restrict__ pooled) {
  int t = blockIdx.x * blockDim.x + threadIdx.x;   // 0 .. N*HW/4 - 1
  int n = t / (PHW / 4);
  int p = (t - n * (PHW / 4)) * 4;
  const float* xb = x + (size_t)n * PCHW + p;

  float sx = 0.f, sy = 0.f, sz = 0.f, sw = 0.f;
  float mx = -INFINITY, my = -INFINITY, mz = -INFINITY, mw = -INFINITY;
#pragma unroll 8
  for (int c = 0; c < PC; ++c) {
    const float4 v = *(const float4*)(xb + (size_t)c * PHW);
    sx += v.x; sy += v.y; sz += v.z; sw += v.w;
    mx = fmaxf(mx, v.x); my = fmaxf(my, v.y);
    mz = fmaxf(mz, v.z); mw = fmaxf(mw, v.w);
  }
  const float inv = 1.0f / (float)PC;
  float* pd = pooled + (size_t)n * 2 * PHW + p;
  float4 a; a.x = sx * inv; a.y = sy * inv; a.z = sz * inv; a.w = sw * inv;
  float4 m; m.x = mx; m.y = my; m.z = mz; m.w = mw;
  *(float4*)(pd)        = a;   // channel 0: mean
  *(float4*)(pd + PHW)  = m;   // channel 1: max
}

// ---------------------------------------------------------------------------
// Kernel 2: 7x7 conv (2 in ch -> 1 out ch, pad 3) + bias + sigmoid via
// V_WMMA_F32_16X16X4_F32 (exact fp32). im2col GEMM: M=16 output pixels per
// wave, K=98 padded to 100 (25 WMMA steps), B = broadcast weights.
// Block = 128 threads = 4 waves = one output row (n,h); wave t covers
// w0 in {0,16,32,40} (overlap writes identical values).
// A layout (16x4 f32): lane L -> M=L&15; VGPR0 holds K=(L>=16?2:0), VGPR1 K+1.
// B layout mirrors A's K striping; value constant over N.
// ---------------------------------------------------------------------------
__global__ void __launch_bounds__(128) sa_conv_sigmoid_kernel(
    const float* __restrict__ pooled,   // [N][2][H][W]
    const float* __restrict__ wconv,    // [2*7*7] = 98 (OIHW, O=1)
    const float* __restrict__ bias,     // [1]
    float* __restrict__ scale) {        // [N][H][W]
  const int nh   = blockIdx.x;          // 0 .. N*H-1
  const int n    = nh / PH;
  const int h    = nh - n * PH;
  const int wave = threadIdx.x >> 5;    // 0..3
  const int lane = threadIdx.x & 31;
  const int w0   = (wave == 3) ? (PW - 16) : (wave * 16);
  const int m    = lane & 15;                 // output pixel within tile
  const int khalf = (lane >> 4) << 1;         // 0 or 2

  const float* pb = pooled + (size_t)n * 2 * PHW;

  v8f acc = {0.f, 0.f, 0.f, 0.f, 0.f, 0.f, 0.f, 0.f};

#pragma unroll
  for (int k0 = 0; k0 < 100; k0 += 4) {
    v2f a, b;
#pragma unroll
    for (int e = 0; e < 2; ++e) {
      const int k = k0 + khalf + e;
      float av = 0.f, bv = 0.f;
      if (k < 98) {
        const int ci  = k / 49;
        const int rem = k - ci * 49;
        const int r   = rem / 7;
        const int s   = rem - r * 7;
        const int hh  = h + r - 3;
        const int ww  = w0 + m + s - 3;
        if (hh >= 0 && hh < PH && ww >= 0 && ww < PW)
          av = pb[ci * PHW + hh * PW + ww];
        bv = wconv[k];
      }
      if (e == 0) { a.x = av; b.x = bv; } else { a.y = av; b.y = bv; }
    }
    // EXEC is all-1s here (no divergence at this point).
    acc = __builtin_amdgcn_wmma_f32_16x16x4_f32(
        /*neg_a=*/false, a, /*neg_b=*/false, b,
        /*c_mod=*/(short)0, acc, /*reuse_a=*/false, /*reuse_b=*/false);
  }

  // D[m][*] is replicated over N (B constant over N). Lanes 0-15 hold rows
  // 0-7 (acc[v] = row v), lanes 16-31 hold rows 8-15 (acc[v] = row v+8).
  // Writers: lanes 0-7 emit pixels 0-7; lanes 16-23 emit pixels 8-15.
  int rr = 0, mout = 0;
  bool writer = false;
  if (lane < 8)                    { writer = true; rr = lane;      mout = lane;      }
  else if (lane >= 16 && lane < 24){ writer = true; rr = lane - 16; mout = lane - 8;  }

  float y = acc[0];
  if (rr == 1) y = acc[1];
  if (rr == 2) y = acc[2];
  if (rr == 3) y = acc[3];
  if (rr == 4) y = acc[4];
  if (rr == 5) y = acc[5];
  if (rr == 6) y = acc[6];
  if (rr == 7) y = acc[7];

  if (writer) {
    y += bias[0];
    const float sg = 1.0f / (1.0f + __expf(-y));
    scale[(size_t)n * PHW + h * PW + w0 + mout] = sg;
  }
}

// ---------------------------------------------------------------------------
// Kernel 3: out = x * scale (broadcast over C). float4 streaming.
// ---------------------------------------------------------------------------
__global__ void __launch_bounds__(256) sa_apply_kernel(
    const float* __restrict__ x, const float* __restrict__ scale,
    float* __restrict__ out) {
  int t = blockIdx.x * blockDim.x + threadIdx.x;   // 0 .. N*C*HW/4 - 1
  int i = t * 4;
  int n = i / PCHW;
  int r = i - n * PCHW;
  int p = r % PHW;                                  // multiple of 4

  const float4 xv = *(const float4*)(x + (size_t)i);
  const float4 sv = *(const float4*)(scale + (size_t)n * PHW + p);
  float4 o;
  o.x = xv.x * sv.x; o.y = xv.y * sv.y;
  o.z = xv.z * sv.z; o.w = xv.w * sv.w;
  *(float4*)(out + (size_t)i) = o;
}

// ---------------------------------------------------------------------------
extern "C" void kernel_launch(void* const* d_in, const int* in_sizes, int n_in,
                              void* d_out, int out_size, void* d_ws, size_t ws_size,
                              hipStream_t stream) {
  const float* x     = (const float*)d_in[0];   // [32,256,56,56]
  const float* wconv = (const float*)d_in[1];   // [1,2,7,7] = 98
  const float* bias  = (const float*)d_in[2];   // [1]
  float* out = (float*)d_out;

  float* pooled = (float*)d_ws;                 // [32,2,56,56] = 200704 floats
  float* scale  = pooled + (size_t)PN * 2 * PHW;// [32,56,56]   = 100352 floats

  // 1) channel mean/max pool: N*HW/4 threads
  sa_pool_kernel<<<(PNHW / 4) / 256, 256, 0, stream>>>(x, pooled);

  // 2) 7x7 conv + sigmoid via WMMA: one block per (n,h) row, 4 waves/row
  sa_conv_sigmoid_kernel<<<PN * PH, 128, 0, stream>>>(pooled, wconv, bias, scale);

  // 3) out = x * scale: N*C*HW/4 threads
  sa_apply_kernel<<<(PN * PCHW / 4) / 256, 256, 0, stream>>>(x, scale, out);
}